// Deform_71897752535328
// MI455X (gfx1250) — compile-verified
//
#include <hip/hip_runtime.h>
#include <stdint.h>
#include <stddef.h>

#define TPB       256
#define PPT       4                   // pixels per thread
#define TILE_PIX  (TPB * PPT)         // 1024 pixels per tile
#define TILE_BYTES (TILE_PIX * 8)     // 8 KB of float2 grid per tile
#define IH 256
#define IW 256
#define MAX_BLOCKS 5632

// Clang-native 4-float vector: valid for __builtin_nontemporal_store and
// lowers to b128 vmem / ds ops.  Supports .x/.y/.z/.w swizzles.
typedef __attribute__((ext_vector_type(4))) float fvec4;

// ---------------------------------------------------------------------------
// Pre-pass: pad RGB (3 floats) -> RGBA (fvec4) so each bilinear tap is one
// aligned global_load_b128. 1 MB result lives in d_ws, stays hot in L2.
// ---------------------------------------------------------------------------
__global__ __launch_bounds__(TPB) void pad_src_kernel(const float* __restrict__ src,
                                                      fvec4* __restrict__ dst,
                                                      int npix)
{
    int i = blockIdx.x * TPB + threadIdx.x;
    if (i < npix) {
        fvec4 v;
        v.x = src[i * 3 + 0];
        v.y = src[i * 3 + 1];
        v.z = src[i * 3 + 2];
        v.w = 0.0f;
        dst[i] = v;
    }
}

// ---------------------------------------------------------------------------
// CDNA5 async global->LDS copy: each thread copies its own 32 bytes.
// INST_OFFSET applies to both the global and LDS addresses, so one VGPR pair
// + one LDS-address VGPR covers both 16B chunks.  Tracked by ASYNCcnt.
// ---------------------------------------------------------------------------
__device__ __forceinline__ void async_copy32(uint32_t lds_addr, uint64_t gaddr)
{
    asm volatile(
        "global_load_async_to_lds_b128 %0, %1, off\n\t"
        "global_load_async_to_lds_b128 %0, %1, off offset:16"
        :
        : "v"(lds_addr), "v"(gaddr)
        : "memory");
}

#define WAIT_ASYNC_LE_2() asm volatile("s_wait_asynccnt 2" ::: "memory")
#define WAIT_ASYNC_LE_0() asm volatile("s_wait_asynccnt 0" ::: "memory")

// ---------------------------------------------------------------------------
// One bilinear sample, torch align_corners=False + zero padding semantics.
//   x = (gx+1)*W/2 - 0.5 = fma(gx, 128, 127.5)
// Branchless OOB: clamp the gather index, zero the corner weight.
// ---------------------------------------------------------------------------
template<bool PADDED>
__device__ __forceinline__ fvec4 sample_one(const fvec4* __restrict__ img4,
                                            const float* __restrict__ img3,
                                            float gx, float gy)
{
    float xf  = fmaf(gx, 128.0f, 127.5f);
    float yf  = fmaf(gy, 128.0f, 127.5f);
    float xwf = floorf(xf);
    float ynf = floorf(yf);
    float fx  = xf - xwf;
    float fy  = yf - ynf;
    int x0 = (int)xwf, y0 = (int)ynf;
    int x1 = x0 + 1,   y1 = y0 + 1;

    bool vx0 = (unsigned)x0 < (unsigned)IW;
    bool vx1 = (unsigned)x1 < (unsigned)IW;
    bool vy0 = (unsigned)y0 < (unsigned)IH;
    bool vy1 = (unsigned)y1 < (unsigned)IH;

    float ex = 1.0f - fx;
    float sy = 1.0f - fy;
    float wnw = (vy0 && vx0) ? sy * ex : 0.0f;
    float wne = (vy0 && vx1) ? sy * fx : 0.0f;
    float wsw = (vy1 && vx0) ? fy * ex : 0.0f;
    float wse = (vy1 && vx1) ? fy * fx : 0.0f;

    int cx0 = x0 < 0 ? 0 : (x0 > IW - 1 ? IW - 1 : x0);
    int cx1 = x1 < 0 ? 0 : (x1 > IW - 1 ? IW - 1 : x1);
    int cy0 = y0 < 0 ? 0 : (y0 > IH - 1 ? IH - 1 : y0);
    int cy1 = y1 < 0 ? 0 : (y1 > IH - 1 ? IH - 1 : y1);

    int inw = cy0 * IW + cx0;
    int ine = cy0 * IW + cx1;
    int isw = cy1 * IW + cx0;
    int ise = cy1 * IW + cx1;

    fvec4 r;
    if constexpr (PADDED) {
        fvec4 pnw = img4[inw];
        fvec4 pne = img4[ine];
        fvec4 psw = img4[isw];
        fvec4 pse = img4[ise];
        r.x = fmaf(wnw, pnw.x, fmaf(wsw, psw.x, fmaf(wne, pne.x, wse * pse.x)));
        r.y = fmaf(wnw, pnw.y, fmaf(wsw, psw.y, fmaf(wne, pne.y, wse * pse.y)));
        r.z = fmaf(wnw, pnw.z, fmaf(wsw, psw.z, fmaf(wne, pne.z, wse * pse.z)));
    } else {
        float acc[3];
#pragma unroll
        for (int k = 0; k < 3; ++k) {
            float pnw = img3[inw * 3 + k];
            float pne = img3[ine * 3 + k];
            float psw = img3[isw * 3 + k];
            float pse = img3[ise * 3 + k];
            acc[k] = fmaf(wnw, pnw, fmaf(wsw, psw, fmaf(wne, pne, wse * pse)));
        }
        r.x = acc[0]; r.y = acc[1]; r.z = acc[2];
    }
    r.w = 0.0f;
    return r;
}

// ---------------------------------------------------------------------------
// Main kernel.  Persistent blocks; each loop iteration handles one 1024-pixel
// tile.  Grid data is double-buffered through LDS with async copies: while
// tile i is gathered/computed/stored, tile i+1 streams in on the async
// engine.  Each wave reads only the LDS bytes its own lanes copied, so
// ordering is purely s_wait_asynccnt (no workgroup barrier needed).
// ---------------------------------------------------------------------------
template<bool PADDED>
__global__ __launch_bounds__(TPB) void deform_main(const float* __restrict__ motions,
                                                   const fvec4* __restrict__ img4,
                                                   const float* __restrict__ img3,
                                                   float*       __restrict__ out,
                                                   int num_tiles)
{
    __shared__ __align__(16) float lds[2 * TILE_PIX * 2];   // 2 x 8 KB

    const int tid = threadIdx.x;
    // Generic pointer to LDS: low 32 bits are the wave-relative LDS byte addr.
    const uint32_t lds_base = (uint32_t)(uintptr_t)(void*)&lds[0];
    const uint32_t lds0 = lds_base + (uint32_t)tid * 32u;
    const uint32_t lds1 = lds0 + (uint32_t)(TILE_PIX * 2 * 4);   // +8192B
    const uint64_t mbase = (uint64_t)(uintptr_t)motions;

    int tile = (int)blockIdx.x;
    const int stride = (int)gridDim.x;
    if (tile >= num_tiles) return;

    // Prologue: start streaming the first tile.
    async_copy32(lds0, mbase + (uint64_t)tile * TILE_BYTES + (uint32_t)tid * 32u);

    int cur = 0;
    while (true) {
        const int nxt = tile + stride;
        if (nxt < num_tiles) {
            // Kick off next tile into the other buffer, then wait until only
            // those 2 async ops are outstanding -> current tile is in LDS.
            async_copy32(cur ? lds0 : lds1,
                         mbase + (uint64_t)nxt * TILE_BYTES + (uint32_t)tid * 32u);
            WAIT_ASYNC_LE_2();
        } else {
            WAIT_ASYNC_LE_0();
        }

        // This thread's 4 grid coords (float2 x4 = 32B) from its own LDS slot.
        const float* myb = &lds[(cur ? (TILE_PIX * 2) : 0) + tid * 8];
        fvec4 g01 = ((const fvec4*)myb)[0];   // gx0 gy0 gx1 gy1
        fvec4 g23 = ((const fvec4*)myb)[1];   // gx2 gy2 gx3 gy3

        fvec4 p0 = sample_one<PADDED>(img4, img3, g01.x, g01.y);
        fvec4 p1 = sample_one<PADDED>(img4, img3, g01.z, g01.w);
        fvec4 p2 = sample_one<PADDED>(img4, img3, g23.x, g23.y);
        fvec4 p3 = sample_one<PADDED>(img4, img3, g23.z, g23.w);

        // 12 output floats, 48B-aligned -> 3 non-temporal b128 stores.
        fvec4 r0; r0.x = p0.x; r0.y = p0.y; r0.z = p0.z; r0.w = p1.x;
        fvec4 r1; r1.x = p1.y; r1.y = p1.z; r1.z = p2.x; r1.w = p2.y;
        fvec4 r2; r2.x = p2.z; r2.y = p3.x; r2.z = p3.y; r2.w = p3.z;

        fvec4* op = (fvec4*)(out + (size_t)tile * (TILE_PIX * 3) + (size_t)tid * (PPT * 3));
        __builtin_nontemporal_store(r0, op + 0);
        __builtin_nontemporal_store(r1, op + 1);
        __builtin_nontemporal_store(r2, op + 2);

        if (nxt >= num_tiles) break;
        tile = nxt;
        cur ^= 1;
    }
}

// ---------------------------------------------------------------------------
// Scalar tail for pixel counts that are not a multiple of TILE_PIX.
// ---------------------------------------------------------------------------
template<bool PADDED>
__global__ __launch_bounds__(TPB) void deform_tail(const float* __restrict__ motions,
                                                   const fvec4* __restrict__ img4,
                                                   const float* __restrict__ img3,
                                                   float*       __restrict__ out,
                                                   int start, int total)
{
    int i = start + blockIdx.x * TPB + threadIdx.x;
    if (i < total) {
        float gx = motions[(size_t)i * 2 + 0];
        float gy = motions[(size_t)i * 2 + 1];
        fvec4 r = sample_one<PADDED>(img4, img3, gx, gy);
        out[(size_t)i * 3 + 0] = r.x;
        out[(size_t)i * 3 + 1] = r.y;
        out[(size_t)i * 3 + 2] = r.z;
    }
}

// ---------------------------------------------------------------------------
extern "C" void kernel_launch(void* const* d_in, const int* in_sizes, int n_in,
                              void* d_out, int out_size, void* d_ws, size_t ws_size,
                              hipStream_t stream)
{
    const float* src     = (const float*)d_in[0];   // (1, 256, 256, 3) float32
    const float* motions = (const float*)d_in[1];   // (16, 11, 256, 256, 2) float32
    float*       out     = (float*)d_out;           // (16, 11, 256, 256, 3) float32

    const int src_pix = in_sizes[0] / 3;                       // 65536
    const long long total_pix = (long long)in_sizes[1] / 2;    // 11,534,336
    const int nt  = (int)(total_pix / TILE_PIX);               // full tiles
    const int rem_start = nt * TILE_PIX;
    const int rem = (int)(total_pix - rem_start);

    const bool padded = ws_size >= (size_t)src_pix * sizeof(fvec4);
    fvec4* img4 = (fvec4*)d_ws;

    if (padded) {
        pad_src_kernel<<<(src_pix + TPB - 1) / TPB, TPB, 0, stream>>>(src, img4, src_pix);
    }

    if (nt > 0) {
        int nb = nt < MAX_BLOCKS ? nt : MAX_BLOCKS;
        if (padded)
            deform_main<true><<<nb, TPB, 0, stream>>>(motions, img4, src, out, nt);
        else
            deform_main<false><<<nb, TPB, 0, stream>>>(motions, img4, src, out, nt);
    }
    if (rem > 0) {
        int nb = (rem + TPB - 1) / TPB;
        if (padded)
            deform_tail<true><<<nb, TPB, 0, stream>>>(motions, img4, src, out,
                                                      rem_start, (int)total_pix);
        else
            deform_tail<false><<<nb, TPB, 0, stream>>>(motions, img4, src, out,
                                                       rem_start, (int)total_pix);
    }
}